// Attention_43215960932503
// MI455X (gfx1250) — compile-verified
//
#include <hip/hip_runtime.h>
#include <hip/hip_bf16.h>

// MI455X / gfx1250, wave32. Fused attention with V_WMMA_F32_16X16X4_F32.
//
// Shapes (fixed by the reference): B=2, N=3136 (=196*16), heads=8, kd=16, d=32,
// DIM=DIM_S=256. Workspace layout (floats):
//   Q  : (b,h,n,16)   802816
//   Kt : (b,h,n,16)   802816
//   V  : (b,256,n)   1605632
//   AO : (b,h,n,32)  1605632   (post-softmax, post-relu attention output)
// total 4,816,896 floats = 18.4 MiB (L2-resident on 192MB L2).

typedef __attribute__((ext_vector_type(2))) float v2f;
typedef __attribute__((ext_vector_type(8))) float v8f;

#define NTOT 3136
#define NT   196   // NTOT / 16

__device__ __forceinline__ v8f wmma4(v2f a, v2f b, v8f c) {
  // D(16x16,f32) = A(16x4,f32) * B(4x16,f32) + C
  return __builtin_amdgcn_wmma_f32_16x16x4_f32(
      /*neg_a=*/false, a, /*neg_b=*/false, b,
      /*c_mod=*/(short)0, c, /*reuse_a=*/false, /*reuse_b=*/false);
}

// ---------------------------------------------------------------------------
// Generic 16x16-output-tile GEMM with folded BN scale/bias.
//   out[o,n] = scale[o] * (sum_c W[o,c] * X[c,n]) + bias[o]
// MODE   0: store row-major (b, Cout, N)
//        1: store head-transposed (b, o/16, n, o%16)   -> Q / Kt layout
// XLAYOUT 0: X is (b, Cin, N) row-major
//         1: X is AO (b, h, n, 32); channel c -> (h=c/32, d=c%32)
// One wave per output tile; 4 waves / 128-thread block.
// ---------------------------------------------------------------------------
template <int MODE, int XLAYOUT>
__global__ void __launch_bounds__(128) proj16(
    const float* __restrict__ W, const float* __restrict__ sc,
    const float* __restrict__ bi, const float* __restrict__ X,
    float* __restrict__ out, int Cin, int Cout)
{
  const int lane = threadIdx.x & 31;
  const int widx = threadIdx.x >> 5;
  const int l16  = lane & 15;
  const int hi   = lane >> 4;

  const int tile     = blockIdx.x * 4 + widx;
  const int tilesPerB = (Cout >> 4) * NT;
  const int b  = tile / tilesPerB;
  const int t  = tile - b * tilesPerB;
  const int ot = t / NT;
  const int nt = t - ot * NT;
  const int n  = nt * 16 + l16;

  const float* Xb = X + (size_t)b * Cin * NTOT;
  const float* Wr = W + (size_t)(ot * 16 + l16) * Cin;  // A: row o per lane

  v8f acc = {0.f, 0.f, 0.f, 0.f, 0.f, 0.f, 0.f, 0.f};

  for (int ct = 0; ct < (Cin >> 4); ++ct) {
#pragma unroll
    for (int c = 0; c < 4; ++c) {
      const int k = 4 * c + 2 * hi;            // K pair {k, k+1} for this lane
      v2f a = *(const v2f*)(Wr + ct * 16 + k); // A(16x4): lane l16 row, K=k,k+1
      v2f bb;                                   // B(4x16): lane l16 col n
      if (XLAYOUT == 0) {
        const float* xp = Xb + (size_t)(ct * 16 + k) * NTOT + n;
        bb.x = xp[0];
        bb.y = xp[NTOT];
      } else {
        const int cc = ct * 16 + k;            // even -> pair stays in one head
        const float* xp = Xb + ((size_t)(cc >> 5) * NTOT + n) * 32 + (cc & 31);
        bb = *(const v2f*)xp;
      }
      acc = wmma4(a, bb, acc);
    }
  }

  // C layout: VGPR r -> row (ot*16 + r + 8*hi), col n (lane)
#pragma unroll
  for (int r = 0; r < 8; ++r) {
    const int o = ot * 16 + r + 8 * hi;
    const float v = acc[r] * sc[o] + bi[o];
    if (MODE == 0) {
      out[((size_t)b * Cout + o) * NTOT + n] = v;
    } else {
      out[(((size_t)b * (Cout >> 4) + (o >> 4)) * NTOT + n) * 16 + (o & 15)] = v;
    }
  }
}

// ---------------------------------------------------------------------------
// Fused flash attention, one wave per (b, h, 16-row q tile).
//   S^T(16kv x 16q) = Kt_tile(A) x Q_tile(B)        (4 wmma, K=kd=16)
//   online softmax per q-column (per lane: 7 max + 1 shfl_xor(16))
//   O^T(32d x 16q) += V^T_tile(A) x P^T_tile(B)     (8 wmma via LDS transpose)
// Epilogue: O / l, relu, store AO (b,h,n,32) coalesced-ish.
// ---------------------------------------------------------------------------
__global__ void __launch_bounds__(128) attn_fused(
    const float* __restrict__ Q, const float* __restrict__ Kt,
    const float* __restrict__ V, float* __restrict__ AO)
{
  __shared__ float plds[4][16 * 18];  // stride 18: conflict-free, 8B aligned

  const int lane = threadIdx.x & 31;
  const int widx = threadIdx.x >> 5;
  const int l16  = lane & 15;
  const int hi   = lane >> 4;

  const int tile = blockIdx.x * 4 + widx;
  const int bh   = tile / NT;          // 0..15
  const int qt   = tile - bh * NT;     // 0..195
  const int b    = bh >> 3;
  const int h    = bh & 7;

  const float* Qb = Q  + ((size_t)bh * NTOT + qt * 16) * 16;
  const float* Kb = Kt + (size_t)bh * NTOT * 16;
  const float* Vb = V  + ((size_t)b * 256 + h * 32) * NTOT;
  float* lds = plds[widx];

  // Q tile as B operand (K=kd, N=n_q): lane = q column, contiguous kd.
  v2f bq[4];
#pragma unroll
  for (int c = 0; c < 4; ++c)
    bq[c] = *(const v2f*)(Qb + l16 * 16 + 4 * c + 2 * hi);

  float m = -3.0e38f;   // running max for this lane's q column
  float l = 0.0f;       // running sum
  v8f o0 = {0.f, 0.f, 0.f, 0.f, 0.f, 0.f, 0.f, 0.f};  // O^T rows d=0..15
  v8f o1 = {0.f, 0.f, 0.f, 0.f, 0.f, 0.f, 0.f, 0.f};  // O^T rows d=16..31

  for (int kt = 0; kt < NT; ++kt) {
    // ---- S^T = Kt_tile * Q : A lane = kv row (contiguous kd) ----
    const float* Kp = Kb + (size_t)(kt * 16 + l16) * 16;
    v8f S = {0.f, 0.f, 0.f, 0.f, 0.f, 0.f, 0.f, 0.f};
#pragma unroll
    for (int c = 0; c < 4; ++c) {
      v2f a = *(const v2f*)(Kp + 4 * c + 2 * hi);
      S = wmma4(a, bq[c], S);
    }

    // ---- online softmax over kv (this lane holds 8 kv rows of its column;
    //      partner lane (xor 16) holds the other 8) ----
    float tmax = S[0];
#pragma unroll
    for (int r = 1; r < 8; ++r) tmax = fmaxf(tmax, S[r]);
    tmax = fmaxf(tmax, __shfl_xor(tmax, 16, 32));
    const float mn    = fmaxf(m, tmax);
    const float alpha = __expf(m - mn);
    float psum = 0.f;
#pragma unroll
    for (int r = 0; r < 8; ++r) {
      const float p = __expf(S[r] - mn);
      S[r] = p;
      psum += p;
    }
    psum += __shfl_xor(psum, 16, 32);
    l = l * alpha + psum;
    m = mn;
#pragma unroll
    for (int r = 0; r < 8; ++r) { o0[r] *= alpha; o1[r] *= alpha; }

    // ---- transpose P^T (C layout) -> B operand via LDS ----
#pragma unroll
    for (int r = 0; r < 8; ++r)
      lds[(r + 8 * hi) * 18 + l16] = S[r];   // element (kv=r+8hi, q=l16)
    __builtin_amdgcn_wave_barrier();

    // ---- O^T += V^T * P^T ----
    const float* Vp = Vb + (size_t)l16 * NTOT + kt * 16;  // A: row d = l16
#pragma unroll
    for (int c = 0; c < 4; ++c) {
      const int k = 4 * c + 2 * hi;
      v2f bp;                                 // B: column q=l16, K=k,k+1
      bp.x = lds[k * 18 + l16];
      bp.y = lds[(k + 1) * 18 + l16];
      v2f a0 = *(const v2f*)(Vp + k);                        // d in 0..15
      v2f a1 = *(const v2f*)(Vp + (size_t)16 * NTOT + k);    // d in 16..31
      o0 = wmma4(a0, bp, o0);
      o1 = wmma4(a1, bp, o1);
    }
    __builtin_amdgcn_wave_barrier();
  }

  // ---- epilogue: normalize, relu, store AO (b,h,n,32) ----
  const float inv = 1.0f / l;
  float* Ap = AO + ((size_t)bh * NTOT + qt * 16 + l16) * 32;
#pragma unroll
  for (int r = 0; r < 8; ++r) {
    const int d = r + 8 * hi;            // O^T C-layout: VGPR r -> row d
    Ap[d]      = fmaxf(o0[r] * inv, 0.f);
    Ap[d + 16] = fmaxf(o1[r] * inv, 0.f);
  }
}

// ---------------------------------------------------------------------------
extern "C" void kernel_launch(void* const* d_in, const int* in_sizes, int n_in,
                              void* d_out, int out_size, void* d_ws, size_t ws_size,
                              hipStream_t stream) {
  (void)in_sizes; (void)n_in; (void)out_size; (void)ws_size;
  const float* x  = (const float*)d_in[0];
  const float* sx = (const float*)d_in[1];
  const float* Wq = (const float*)d_in[2];
  const float* sq = (const float*)d_in[3];
  const float* bq = (const float*)d_in[4];
  const float* Wk = (const float*)d_in[5];
  const float* sk = (const float*)d_in[6];
  const float* bk = (const float*)d_in[7];
  const float* Wv = (const float*)d_in[8];
  const float* sv = (const float*)d_in[9];
  const float* bv = (const float*)d_in[10];
  const float* Wp = (const float*)d_in[11];
  const float* sp = (const float*)d_in[12];
  const float* bp = (const float*)d_in[13];
  float* out = (float*)d_out;

  float* ws = (float*)d_ws;
  float* Qw = ws;               // 2*8*3136*16
  float* Kw = Qw + 802816;      // 2*8*3136*16
  float* Vw = Kw + 802816;      // 2*256*3136
  float* Aw = Vw + 1605632;     // 2*8*3136*32

  // tiles: q/k: 2*8*196 = 3136 -> 784 blocks * 4 waves (exact)
  //        v/p: 2*16*196 = 6272 -> 1568 blocks (exact)
  proj16<1, 0><<<784, 128, 0, stream>>>(Wq, sq, bq, sx, Qw, 256, 128);
  proj16<1, 0><<<784, 128, 0, stream>>>(Wk, sk, bk, x,  Kw, 256, 128);
  proj16<0, 0><<<1568, 128, 0, stream>>>(Wv, sv, bv, x, Vw, 256, 256);

  attn_fused<<<784, 128, 0, stream>>>(Qw, Kw, Vw, Aw);   // 16 bh * 196 qtiles

  proj16<0, 1><<<1568, 128, 0, stream>>>(Wp, sp, bp, Aw, out, 256, 256);
}